// QuantumCircuitModule_18631568130835
// MI455X (gfx1250) — compile-verified
//
#include <hip/hip_runtime.h>
#include <hip/hip_bf16.h>

// Quantum "quanvolution" layer, reduced analytically:
//  - CRZ gates are diagonal -> phases cancel in |psi|^2 -> no effect.
//  - Product state -> <Z_q> = |v0_q|^2 - |v1_q|^2, closed form in sin/cos of
//    the 4 raw patch angles (full angles, not half) and the RX weight.
// Memory-bound problem (~20 MB traffic -> ~1us at 23.3 TB/s). CDNA5 features
// used: async global->LDS DMA staging (ASYNCcnt path), wave32 LDS sharing of
// per-pixel sincos (each pixel reused by up to 4 outputs in this row tile).

#define HDIM 128
#define WDIM 128
#define HO   125
#define WO   125
#define SROW 136   // LDS row stride in words: 136 % 64 == 8 -> bank = 8*q + (lane>>2), conflict-free

__global__ __launch_bounds__(256) void qcirc_kernel(
    const float* __restrict__ x,     // [B,1,128,128]
    const float* __restrict__ w,     // [1,7], first 4 are RX angles
    float* __restrict__ out)         // [B,4,125,125] (flat reinterpret of [B,P,4])
{
    __shared__ float raw[512];         // 4 rows x 128 cols, staged by async DMA
    __shared__ float sv[4 * SROW];     // sin of pixels, padded rows
    __shared__ float cv[4 * SROW];     // cos of pixels, padded rows

    const int blk = blockIdx.x;
    const int b   = blk / HO;
    const int i   = blk - b * HO;      // patch row
    const int tid = threadIdx.x;

    const float* xrow = x + (size_t)(b * HDIM + i) * WDIM;

    // ---- Stage 4x128 pixels global -> LDS via CDNA5 async DMA (no VGPR roundtrip)
#pragma unroll
    for (int k = 0; k < 2; ++k) {
        int t = tid + k * 256;                      // 0..511
        const float* gp = xrow + (t >> 7) * WDIM + (t & (WDIM - 1));
        unsigned long long ga = (unsigned long long)(size_t)gp;
        unsigned lo = (unsigned)(size_t)&raw[t];    // low 32 bits = LDS byte address
        asm volatile("global_load_async_to_lds_b32 %0, %1, off"
                     :: "v"(lo), "v"(ga) : "memory");
    }
    asm volatile("s_wait_asynccnt 0x0" ::: "memory");
    __syncthreads();

    // ---- One sincos per pixel, shared across the 4 outputs that reuse it
#pragma unroll
    for (int k = 0; k < 2; ++k) {
        int t   = tid + k * 256;
        int row = t >> 7, col = t & (WDIM - 1);
        float s, c;
        __sincosf(raw[t], &s, &c);                  // v_sin_f32 / v_cos_f32 (TRANS, co-exec)
        sv[row * SROW + col] = s;
        cv[row * SROW + col] = c;
    }
    __syncthreads();

    // ---- 500 outputs per block: o = j*4 + q, contiguous chunk out[b][i*500 + o]
    const int q = tid & 3;                          // fixed per thread (stride 256 % 4 == 0)
    float sw, cw;
    __sincosf(w[q], &sw, &cw);

    float* orow = out + (size_t)b * (4 * HO * WO) + (size_t)i * (4 * WO);
#pragma unroll
    for (int k = 0; k < 2; ++k) {
        int o = tid + k * 256;
        if (o < 4 * WO) {
            int j    = o >> 2;
            int base = q * SROW + j;
            float s0 = sv[base + 0], c0 = cv[base + 0];
            float s1 = sv[base + 1], c1 = cv[base + 1];
            float s2 = sv[base + 2], c2 = cv[base + 2];
            float s3 = sv[base + 3], c3 = cv[base + 3];
            float c01 = c0 * c1;
            float t1 = fmaf(c01, s2, s0 * c2);          // s0*c2 + c0*c1*s2
            float t2 = fmaf(c01, c2, -(s0 * s2));       // c0*c1*c2 - s0*s2
            float t3 = fmaf(c0 * s1, c3, s3 * t2);      // c0*s1*c3 + s3*t2
            orow[o]  = fmaf(sw, t3, -(cw * t1));        // -cos(w)*t1 + sin(w)*t3
        }
    }
}

extern "C" void kernel_launch(void* const* d_in, const int* in_sizes, int n_in,
                              void* d_out, int out_size, void* d_ws, size_t ws_size,
                              hipStream_t stream) {
    const float* x = (const float*)d_in[0];   // [B,1,128,128] fp32
    const float* w = (const float*)d_in[1];   // [1,7] fp32
    float* out     = (float*)d_out;           // [B,4,125,125] fp32

    const int B = in_sizes[0] / (HDIM * WDIM);
    qcirc_kernel<<<dim3(B * HO), dim3(256), 0, stream>>>(x, w, out);
}